// GATLayer_49177375539798
// MI455X (gfx1250) — compile-verified
//
#include <hip/hip_runtime.h>

// GAT layer fused for gfx1250 (MI455X): flash-style attention, fp32 WMMA 16x16x4,
// TDM (tensor_load_to_lds) double-buffered staging of the H tile.
// B=8, N=2048, D=128.

#define B_ 8
#define N_ 2048
#define D_ 128
#define ALPHA_ 0.2f
#define NEGINF_ (-10000.0f)
#define EPS_ 1e-5f

typedef float v2f __attribute__((ext_vector_type(2)));
typedef float v8f __attribute__((ext_vector_type(8)));
typedef unsigned int u32;
typedef u32 u32x4 __attribute__((ext_vector_type(4)));
typedef u32 u32x8 __attribute__((ext_vector_type(8)));

__device__ __forceinline__ v8f wmma_f32(v2f a, v2f b, v8f c) {
    // D = A(16x4) * B(4x16) + C(16x16), all fp32.
    return __builtin_amdgcn_wmma_f32_16x16x4_f32(false, a, false, b, (short)0, c,
                                                 false, false);
}

__device__ __forceinline__ v8f vzero8() {
    v8f z = {0.f, 0.f, 0.f, 0.f, 0.f, 0.f, 0.f, 0.f};
    return z;
}

// TDM: DMA a 64-row x 128-float tile (rows contiguous, row stride 128 floats in
// memory) into LDS at lds_off, inserting 8 dwords of padding after every 128
// dwords -> LDS row stride 136 floats. 2-group D# (2D tensor), wave-level op.
__device__ __forceinline__ void tdm_load_tile(u32 lds_off, unsigned long long gaddr) {
    u32x4 g0;
    g0[0] = 1u;                                   // count=1, user descriptor
    g0[1] = lds_off;                              // LDS byte address
    g0[2] = (u32)(gaddr & 0xffffffffull);         // global_addr[31:0]
    g0[3] = (u32)((gaddr >> 32) & 0x01ffffffull)  // global_addr[56:32]
            | 0x80000000u;                        // type=2 ("image")
    u32x8 g1;
    g1[0] = (2u << 16)      // data_size = 4B
          | (1u << 20)      // pad_enable
          | (6u << 22)      // pad_interval: 128 dwords
          | (7u << 25);     // pad_amount: 8 dwords
    g1[1] = (128u << 16);   // [15:0] abar=0, [31:16] tensor_dim0 lo = 128
    g1[2] = (64u << 16);    // [15:0] tensor_dim0 hi, [31:16] tensor_dim1 lo = 64
    g1[3] = (128u << 16);   // [15:0] tensor_dim1 hi, [31:16] tile_dim0 = 128
    g1[4] = 64u;            // [15:0] tile_dim1 = 64, [31:16] tile_dim2 = 0
    g1[5] = 128u;           // tensor_dim0_stride[31:0] = 128 elements
    g1[6] = 0u;             // stride hi / tensor_dim1_stride lo
    g1[7] = 0u;             // tensor_dim1_stride hi
    asm volatile("tensor_load_to_lds %0, %1" :: "s"(g0), "s"(g1) : "memory");
}

// ---------------------------------------------------------------------------
// Kernel 1: h = (x * node_mask) @ W ; e_l = h @ a_l ; e_r = h @ a_r
// Grid: (B*N)/64 blocks of 128 threads (4 waves). Wave owns a 16-row tile.
// ---------------------------------------------------------------------------
__global__ __launch_bounds__(128)
void gat_h_kernel(const float* __restrict__ x, const int* __restrict__ node_mask,
                  const float* __restrict__ W, const float* __restrict__ a_l,
                  const float* __restrict__ a_r, float* __restrict__ h,
                  float* __restrict__ e_l, float* __restrict__ e_r) {
    __shared__ float w_lds[D_ * D_];   // W[k][n], 64 KB

    const int tid = threadIdx.x;
    {   // cooperative W load (float4)
        const float4* Wv = (const float4*)W;
        float4* Lv = (float4*)w_lds;
        #pragma unroll
        for (int i = 0; i < (D_ * D_ / 4) / 128; ++i)
            Lv[tid + i * 128] = Wv[tid + i * 128];
    }
    __syncthreads();

    const int wave = tid >> 5;
    const int lane = tid & 31;
    const int lo = lane & 15;
    const int hi = lane >> 4;
    const int row_base = blockIdx.x * 64 + wave * 16;  // flattened [B*N) row
    const int arow = row_base + lo;                    // row this lane feeds (A frag)
    const float mrow = (float)node_mask[arow];
    const float* xr = x + (size_t)arow * D_;

    v8f acc[8];
    #pragma unroll
    for (int t = 0; t < 8; ++t) acc[t] = vzero8();

    #pragma unroll 4
    for (int kk = 0; kk < D_ / 4; ++kk) {
        const int kb = kk * 4 + hi * 2;
        v2f a;
        a.x = xr[kb] * mrow;
        a.y = xr[kb + 1] * mrow;
        #pragma unroll
        for (int t = 0; t < 8; ++t) {
            v2f b;
            b.x = w_lds[kb * D_ + t * 16 + lo];
            b.y = w_lds[(kb + 1) * D_ + t * 16 + lo];
            acc[t] = wmma_f32(a, b, acc[t]);
        }
    }

    // Store h and fold in e_l/e_r partial dot-products.
    float pl[8], pr[8];
    #pragma unroll
    for (int v = 0; v < 8; ++v) { pl[v] = 0.f; pr[v] = 0.f; }

    #pragma unroll
    for (int t = 0; t < 8; ++t) {
        const int col = t * 16 + lo;
        const float alc = a_l[col];
        const float arc = a_r[col];
        #pragma unroll
        for (int v = 0; v < 8; ++v) {
            const int r = row_base + hi * 8 + v;   // C/D layout row
            const float hv = acc[t][v];
            h[(size_t)r * D_ + col] = hv;
            pl[v] += hv * alc;
            pr[v] += hv * arc;
        }
    }
    #pragma unroll
    for (int m = 1; m < 16; m <<= 1) {
        #pragma unroll
        for (int v = 0; v < 8; ++v) {
            pl[v] += __shfl_xor(pl[v], m, 32);
            pr[v] += __shfl_xor(pr[v], m, 32);
        }
    }
    if (lo == 0) {
        #pragma unroll
        for (int v = 0; v < 8; ++v) {
            const int r = row_base + hi * 8 + v;
            e_l[r] = pl[v];
            e_r[r] = pr[v];
        }
    }
}

// ---------------------------------------------------------------------------
// Kernel 2: fused masked-LeakyReLU logits -> online softmax -> attn @ h
//           -> residual -> node mask -> LayerNorm.
// Grid: B*(N/64) blocks of 128 threads. Wave owns 16 i-rows; block streams
// j in 64-wide chunks; the H tile is DMA'd into double-buffered LDS by the
// Tensor Data Mover (overlapped with compute). adj is read exactly once.
// ---------------------------------------------------------------------------
__global__ __launch_bounds__(128)
void gat_attn_kernel(const float* __restrict__ x, const int* __restrict__ adj,
                     const int* __restrict__ node_mask,
                     const float* __restrict__ h, const float* __restrict__ e_l,
                     const float* __restrict__ e_r,
                     const float* __restrict__ gamma, const float* __restrict__ beta,
                     float* __restrict__ out) {
    constexpr int HS = 136;                  // padded LDS row stride (TDM pad)
    constexpr int BUF = 64 * HS;             // floats per buffer
    __shared__ float h_lds[2 * BUF];         // double-buffered H tile (~68 KB)
    __shared__ float er_lds[64];
    __shared__ int   mj_lds[64];

    const int tid = threadIdx.x;
    const int wave = tid >> 5;
    const int lane = tid & 31;
    const int lo = lane & 15;
    const int hi = lane >> 4;
    const int b  = blockIdx.x >> 5;          // 32 i-blocks per batch
    const int i0 = (blockIdx.x & 31) * 64;
    const int iw0 = i0 + wave * 16;
    const int g_ia = b * N_ + iw0 + lo;      // row this lane feeds (A/P frag)
    const int mask_i = node_mask[g_ia];
    const float eli = e_l[g_ia];
    const int* adj_row = adj + (size_t)g_ia * N_;

    const u32 lds_base = (u32)(uintptr_t)(&h_lds[0]);   // LDS byte offset
    const unsigned long long h_base =
        (unsigned long long)(uintptr_t)(h + ((size_t)b * N_) * D_);

    v8f acc[8];
    #pragma unroll
    for (int t = 0; t < 8; ++t) acc[t] = vzero8();
    float m_run = -3.0e38f;
    float l_run = 0.f;

    // Prologue: prefetch chunk-0 e_r/mask into regs, kick off chunk-0 DMA.
    float er_nx = 0.f;
    int   mj_nx = 0;
    if (tid < 64) {
        er_nx = e_r[b * N_ + tid];
        mj_nx = node_mask[b * N_ + tid];
    }
    if (wave == 0) tdm_load_tile(lds_base, h_base);

    for (int jc = 0; jc < N_; jc += 64) {
        const int cur = (jc >> 6) & 1;
        // Publish this chunk's e_r/mask (prev compute finished at bottom barrier).
        if (tid < 64) { er_lds[tid] = er_nx; mj_lds[tid] = mj_nx; }
        // Kick off next chunk's DMA into the other buffer, then wait for ours.
        if (wave == 0) {
            if (jc + 64 < N_) {
                tdm_load_tile(lds_base + (u32)(cur ^ 1) * (BUF * 4),
                              h_base + (unsigned long long)(jc + 64) * (D_ * 4));
                __builtin_amdgcn_s_wait_tensorcnt(1);   // ours (older) is done
            } else {
                __builtin_amdgcn_s_wait_tensorcnt(0);
            }
        }
        // Prefetch next chunk's e_r/mask into regs; hint next adj stretch.
        if (jc + 64 < N_) {
            if (tid < 64) {
                er_nx = e_r[b * N_ + jc + 64 + tid];
                mj_nx = node_mask[b * N_ + jc + 64 + tid];
            }
            __builtin_prefetch(adj_row + jc + 64, 0, 0);  // global_prefetch_b8
        }
        __syncthreads();    // H tile + er/mj visible to all waves

        const float* hb = &h_lds[cur * BUF];

        // Masked LeakyReLU logits, 32 per lane, directly in A-fragment order.
        float p[32];
        float lmax = -3.0e38f;
        #pragma unroll
        for (int kk = 0; kk < 16; ++kk) {
            const int kb = kk * 4 + hi * 2;
            const int2 av = *(const int2*)(adj_row + jc + kb);   // 8B aligned
            #pragma unroll
            for (int jj = 0; jj < 2; ++jj) {
                const int j = kb + jj;
                const int aok = (jj == 0) ? av.x : av.y;
                float s = eli + er_lds[j];
                s = (s > 0.f) ? s : ALPHA_ * s;
                const bool ok = (aok != 0) & (mask_i != 0) & (mj_lds[j] != 0);
                s = ok ? s : NEGINF_;
                p[kk * 2 + jj] = s;
                lmax = fmaxf(lmax, s);
            }
        }
        lmax = fmaxf(lmax, __shfl_xor(lmax, 16, 32));   // pair lanes share a row
        const float m_new = fmaxf(m_run, lmax);
        const float scale = __expf(m_run - m_new);

        float lsum = 0.f;
        #pragma unroll
        for (int q = 0; q < 32; ++q) {
            p[q] = __expf(p[q] - m_new);
            lsum += p[q];
        }
        lsum += __shfl_xor(lsum, 16, 32);
        l_run = l_run * scale + lsum;
        m_run = m_new;

        // Rescale accumulators: scale for row r lives on lane r (r = v + 8*hi).
        #pragma unroll
        for (int v = 0; v < 8; ++v) {
            const float sv = __shfl(scale, v + hi * 8, 32);
            #pragma unroll
            for (int t = 0; t < 8; ++t) acc[t][v] *= sv;
        }

        // P(16x64) @ H(64x128) via 16 k-steps of WMMA f32 16x16x4, 8 col tiles.
        #pragma unroll
        for (int kk = 0; kk < 16; ++kk) {
            const int kb = kk * 4 + hi * 2;
            v2f a;
            a.x = p[kk * 2];
            a.y = p[kk * 2 + 1];
            #pragma unroll
            for (int t = 0; t < 8; ++t) {
                v2f bf;
                bf.x = hb[kb * HS + t * 16 + lo];
                bf.y = hb[(kb + 1) * HS + t * 16 + lo];
                acc[t] = wmma_f32(a, bf, acc[t]);
            }
        }
        __syncthreads();    // all waves done with buffer `cur` before reuse
    }

    // Epilogue: O/l, residual + node mask, LayerNorm over D, store.
    const float linv = 1.0f / l_run;
    #pragma unroll
    for (int v = 0; v < 8; ++v) {
        const float sv = __shfl(linv, v + hi * 8, 32);
        const int g_r = b * N_ + iw0 + hi * 8 + v;
        const float mr = (float)node_mask[g_r];
        const float* xr = x + (size_t)g_r * D_;
        float vals[8];
        float sum = 0.f, sq = 0.f;
        #pragma unroll
        for (int t = 0; t < 8; ++t) {
            const int col = t * 16 + lo;
            float o = acc[t][v] * sv;
            o = (o + xr[col] * mr) * mr;    // x pre-masked => (o + x*m)*m
            vals[t] = o;
            sum += o;
            sq += o * o;
        }
        #pragma unroll
        for (int m = 1; m < 16; m <<= 1) {  // 16 lanes (same hi) share this row
            sum += __shfl_xor(sum, m, 32);
            sq  += __shfl_xor(sq,  m, 32);
        }
        const float mean = sum * (1.0f / 128.0f);
        const float var  = sq * (1.0f / 128.0f) - mean * mean;
        const float inv  = rsqrtf(var + EPS_);
        #pragma unroll
        for (int t = 0; t < 8; ++t) {
            const int col = t * 16 + lo;
            out[(size_t)g_r * D_ + col] = (vals[t] - mean) * inv * gamma[col] + beta[col];
        }
    }
}

// ---------------------------------------------------------------------------
extern "C" void kernel_launch(void* const* d_in, const int* in_sizes, int n_in,
                              void* d_out, int out_size, void* d_ws, size_t ws_size,
                              hipStream_t stream) {
    const float* x         = (const float*)d_in[0];
    const int*   adj       = (const int*)d_in[1];
    const int*   node_mask = (const int*)d_in[2];
    const float* W         = (const float*)d_in[3];
    const float* a_l       = (const float*)d_in[4];
    const float* a_r       = (const float*)d_in[5];
    const float* gamma     = (const float*)d_in[6];
    const float* beta      = (const float*)d_in[7];
    float* out = (float*)d_out;

    // workspace: h [B*N*D] | e_l [B*N] | e_r [B*N]
    float* h   = (float*)d_ws;
    float* e_l = h + (size_t)B_ * N_ * D_;
    float* e_r = e_l + (size_t)B_ * N_;

    gat_h_kernel<<<(B_ * N_) / 64, 128, 0, stream>>>(x, node_mask, W, a_l, a_r,
                                                     h, e_l, e_r);
    gat_attn_kernel<<<(B_ * N_) / 64, 128, 0, stream>>>(x, adj, node_mask, h,
                                                        e_l, e_r, gamma, beta, out);
}